// HypersphereContrastiveLoss_85555748537000
// MI455X (gfx1250) — compile-verified
//
#include <hip/hip_runtime.h>
#include <math.h>

// ---------------------------------------------------------------------------
// Symmetric contrastive (CLIP/InfoNCE) loss, B=16384, D=640, fp32.
// sim = clip((a_n . b_n)/T, +-20). Clamp bound => softmax needs no max pass:
// stream per-row/per-col {sum exp, sum, packed argmax, diag} while doing a
// WMMA f32 16x16x4 GEMM. Both inputs fit in MI455X's 192MB L2 -> compute bound.
// ---------------------------------------------------------------------------

#define B_ROWS 16384
#define D_DIM  640
#define INV_T  (1.0f / 0.07f)
#define CLAMP_V 20.0f

#define TM 128   // rows per workgroup tile
#define TN 256   // cols per workgroup tile
#define KB 32    // k-block
#define KS 36    // padded LDS row stride (floats): 16B-aligned, conflict-free

typedef __attribute__((ext_vector_type(2))) float v2f;
typedef __attribute__((ext_vector_type(8))) float v8f;

__device__ __forceinline__ unsigned int orderable_f32(float f) {
  unsigned int u = __float_as_uint(f);
  return (u & 0x80000000u) ? ~u : (u | 0x80000000u);
}
// Lexicographic (value, first-index) max key: larger value wins; on ties the
// smaller index wins (matches jnp.argmax) because we store ~idx in the low half.
__device__ __forceinline__ unsigned long long pack_max(float s, unsigned int idx) {
  return ((unsigned long long)orderable_f32(s) << 32) |
         (unsigned long long)(~idx);
}
__device__ __forceinline__ unsigned long long umax64(unsigned long long a, unsigned long long b) {
  return a > b ? a : b;
}
__device__ __forceinline__ unsigned long long shfl_xor_u64(unsigned long long v, int m) {
  unsigned int lo = (unsigned int)v;
  unsigned int hi = (unsigned int)(v >> 32);
  lo = (unsigned int)__shfl_xor((int)lo, m);
  hi = (unsigned int)__shfl_xor((int)hi, m);
  return ((unsigned long long)hi << 32) | (unsigned long long)lo;
}

// ---------------------------------------------------------------------------
// Kernel 1: zero the global accumulators (ws is poisoned by the harness).
// ---------------------------------------------------------------------------
__global__ void init_acc_kernel(float* __restrict__ rowExp, float* __restrict__ rowSum,
                                float* __restrict__ colExp,
                                unsigned long long* __restrict__ rowMax,
                                unsigned long long* __restrict__ colMax) {
  int i = blockIdx.x * blockDim.x + threadIdx.x;
  if (i < B_ROWS) {
    rowExp[i] = 0.0f;
    rowSum[i] = 0.0f;
    colExp[i] = 0.0f;
    rowMax[i] = 0ull;   // < pack of any real value
    colMax[i] = 0ull;
  }
}

// ---------------------------------------------------------------------------
// Kernel 2: inverse L2 norms (one block per row; y-dim selects a or b).
// ---------------------------------------------------------------------------
__global__ __launch_bounds__(256)
void inv_norm_kernel(const float* __restrict__ a, const float* __restrict__ b,
                     float* __restrict__ inva, float* __restrict__ invb) {
  const float* src = blockIdx.y ? b : a;
  float* dst       = blockIdx.y ? invb : inva;
  const int row = blockIdx.x;
  float ss = 0.0f;
  for (int d = threadIdx.x; d < D_DIM; d += 256) {
    float v = src[(size_t)row * D_DIM + d];
    ss += v * v;
  }
#pragma unroll
  for (int m = 16; m >= 1; m >>= 1) ss += __shfl_xor(ss, m);
  __shared__ float scratch[8];
  if ((threadIdx.x & 31) == 0) scratch[threadIdx.x >> 5] = ss;
  __syncthreads();
  if (threadIdx.x == 0) {
    float t = 0.0f;
#pragma unroll
    for (int i = 0; i < 8; ++i) t += scratch[i];
    dst[row] = 1.0f / fmaxf(sqrtf(t), 1e-12f);
  }
}

// ---------------------------------------------------------------------------
// Kernel 3: WMMA GEMM tile (128x256) with fused softmax/metric reductions.
// 8 waves, each owns a 64x64 sub-tile = 4x4 grid of 16x16x4 f32 WMMAs.
// ---------------------------------------------------------------------------
__global__ __launch_bounds__(256)
void sim_gemm_reduce_kernel(const float* __restrict__ A, const float* __restrict__ Bm,
                            const float* __restrict__ inva, const float* __restrict__ invb,
                            float* __restrict__ rowExp, float* __restrict__ rowSum,
                            float* __restrict__ colExp,
                            unsigned long long* __restrict__ rowMax,
                            unsigned long long* __restrict__ colMax,
                            float* __restrict__ diagOut) {
  __shared__ __align__(16) float As[TM * KS];
  __shared__ __align__(16) float Bs[TN * KS];
  __shared__ float ldsRowE[TM], ldsRowS[TM];
  __shared__ unsigned long long ldsRowM[TM];
  __shared__ float ldsColE[TN];
  __shared__ unsigned long long ldsColM[TN];
  __shared__ float rowScale[TM], colScale[TN];

  const int t     = threadIdx.x;
  const int tileM = blockIdx.y * TM;
  const int tileN = blockIdx.x * TN;

  if (t < TM) {
    ldsRowE[t] = 0.0f; ldsRowS[t] = 0.0f; ldsRowM[t] = 0ull;
    rowScale[t] = inva[tileM + t];
  }
  ldsColE[t] = 0.0f; ldsColM[t] = 0ull;
  colScale[t] = invb[tileN + t];

  const int lane = t & 31;
  const int l16  = lane & 15;
  const int kh   = lane >> 4;         // which K-half of the wave
  const int w    = t >> 5;
  const int wm   = (w >> 2) * 64;     // wave row offset within tile (2 rows of waves)
  const int wn   = (w & 3) * 64;      // wave col offset within tile (4 cols of waves)

  v8f acc[4][4];
#pragma unroll
  for (int mi = 0; mi < 4; ++mi)
#pragma unroll
    for (int ni = 0; ni < 4; ++ni)
#pragma unroll
      for (int e = 0; e < 8; ++e) acc[mi][ni][e] = 0.0f;

  for (int kb = 0; kb < D_DIM / KB; ++kb) {
    const int k0 = kb * KB;
    // Stage A tile (128x32) and B tile (256x32, row j = embedding b_j) in LDS.
#pragma unroll
    for (int p = 0; p < 4; ++p) {
      int q = t + p * 256;
      int r = q >> 3, c = (q & 7) << 2;
      *(float4*)&As[r * KS + c] =
          *(const float4*)&A[(size_t)(tileM + r) * D_DIM + k0 + c];
    }
#pragma unroll
    for (int p = 0; p < 8; ++p) {
      int q = t + p * 256;
      int r = q >> 3, c = (q & 7) << 2;
      *(float4*)&Bs[r * KS + c] =
          *(const float4*)&Bm[(size_t)(tileN + r) * D_DIM + k0 + c];
    }
    // Prefetch next K-block (global_prefetch_b8 on gfx1250).
    if (kb + 1 < D_DIM / KB) {
      const int kn = k0 + KB;
      if (t < TM) __builtin_prefetch(&A[(size_t)(tileM + t) * D_DIM + kn], 0, 0);
      __builtin_prefetch(&Bm[(size_t)(tileN + t) * D_DIM + kn], 0, 0);
    }
    __syncthreads();

#pragma unroll
    for (int kk = 0; kk < KB / 4; ++kk) {
      v2f af[4], bf[4];
#pragma unroll
      for (int mi = 0; mi < 4; ++mi)
        af[mi] = *(const v2f*)&As[(wm + mi * 16 + l16) * KS + kk * 4 + 2 * kh];
#pragma unroll
      for (int ni = 0; ni < 4; ++ni)
        bf[ni] = *(const v2f*)&Bs[(wn + ni * 16 + l16) * KS + kk * 4 + 2 * kh];
#pragma unroll
      for (int mi = 0; mi < 4; ++mi)
#pragma unroll
        for (int ni = 0; ni < 4; ++ni)
          acc[mi][ni] = __builtin_amdgcn_wmma_f32_16x16x4_f32(
              false, af[mi], false, bf[ni], (short)0, acc[mi][ni], false, false);
    }
    __syncthreads();
  }

  // ---------------- Epilogue: scale, clamp, and streaming reductions -------
  float colE[4];
  unsigned long long colM[4];
#pragma unroll
  for (int ni = 0; ni < 4; ++ni) { colE[ni] = 0.0f; colM[ni] = 0ull; }

#pragma unroll
  for (int mi = 0; mi < 4; ++mi) {
#pragma unroll
    for (int v = 0; v < 8; ++v) {
      const int r  = wm + mi * 16 + v + 8 * kh;   // C layout: lanes>=16 hold M+8
      const int gi = tileM + r;
      const float ra = rowScale[r];
      float rowE = 0.0f, rowS = 0.0f;
      unsigned long long rowM = 0ull;
#pragma unroll
      for (int ni = 0; ni < 4; ++ni) {
        const int c  = wn + ni * 16 + l16;
        const int gj = tileN + c;
        float s = acc[mi][ni][v] * ra * colScale[c] * INV_T;
        s = fminf(CLAMP_V, fmaxf(-CLAMP_V, s));
        const float e = __expf(s);
        rowE += e; rowS += s;
        rowM = umax64(rowM, pack_max(s, (unsigned)gj));
        colE[ni] += e;
        colM[ni] = umax64(colM[ni], pack_max(s, (unsigned)gi));
        if (gi == gj) diagOut[gi] = s;   // each diag element computed once
      }
      // Reduce across the 16 lanes sharing this row (xor masks stay in-half).
#pragma unroll
      for (int m = 8; m >= 1; m >>= 1) {
        rowE += __shfl_xor(rowE, m);
        rowS += __shfl_xor(rowS, m);
        rowM = umax64(rowM, shfl_xor_u64(rowM, m));
      }
      if (l16 == 0) {
        atomicAdd(&ldsRowE[r], rowE);
        atomicAdd(&ldsRowS[r], rowS);
        atomicMax(&ldsRowM[r], rowM);
      }
    }
  }
#pragma unroll
  for (int ni = 0; ni < 4; ++ni) {
    const float ce = colE[ni] + __shfl_xor(colE[ni], 16);
    const unsigned long long cm = umax64(colM[ni], shfl_xor_u64(colM[ni], 16));
    if (lane < 16) {
      const int c = wn + ni * 16 + l16;
      atomicAdd(&ldsColE[c], ce);
      atomicMax(&ldsColM[c], cm);
    }
  }
  __syncthreads();

  // One global atomic per row/col per tile.
  if (t < TM) {
    atomicAdd(&rowExp[tileM + t], ldsRowE[t]);
    atomicAdd(&rowSum[tileM + t], ldsRowS[t]);
    atomicMax(&rowMax[tileM + t], ldsRowM[t]);
  }
  atomicAdd(&colExp[tileN + t], ldsColE[t]);
  atomicMax(&colMax[tileN + t], ldsColM[t]);
}

// ---------------------------------------------------------------------------
// Kernel 4: finalize the four scalars.
// ---------------------------------------------------------------------------
__device__ __forceinline__ float block_reduce_sum(float v, float* scratch) {
#pragma unroll
  for (int m = 16; m >= 1; m >>= 1) v += __shfl_xor(v, m);
  __syncthreads();
  if ((threadIdx.x & 31) == 0) scratch[threadIdx.x >> 5] = v;
  __syncthreads();
  float r = (threadIdx.x < 8) ? scratch[threadIdx.x] : 0.0f;
#pragma unroll
  for (int m = 4; m >= 1; m >>= 1) r += __shfl_xor(r, m);
  return r;  // valid on thread 0
}

__global__ __launch_bounds__(256)
void finalize_kernel(const float* __restrict__ rowExp, const float* __restrict__ rowSum,
                     const float* __restrict__ colExp,
                     const unsigned long long* __restrict__ rowMax,
                     const unsigned long long* __restrict__ colMax,
                     const float* __restrict__ diagv, float* __restrict__ out) {
  __shared__ float scratch[8];
  float lsum = 0.0f, psum = 0.0f, nsum = 0.0f, accA = 0.0f, accB = 0.0f;
  for (int i = threadIdx.x; i < B_ROWS; i += 256) {
    const float d = diagv[i];
    lsum += logf(rowExp[i]) + logf(colExp[i]) - 2.0f * d;
    psum += d;
    nsum += rowSum[i] - d;
    const unsigned int ja = ~(unsigned int)(rowMax[i] & 0xFFFFFFFFull);
    const unsigned int ia = ~(unsigned int)(colMax[i] & 0xFFFFFFFFull);
    accA += (ja == (unsigned)i) ? 1.0f : 0.0f;
    accB += (ia == (unsigned)i) ? 1.0f : 0.0f;
  }
  lsum = block_reduce_sum(lsum, scratch);
  psum = block_reduce_sum(psum, scratch);
  nsum = block_reduce_sum(nsum, scratch);
  accA = block_reduce_sum(accA, scratch);
  accB = block_reduce_sum(accB, scratch);
  if (threadIdx.x == 0) {
    const float Bf = (float)B_ROWS;
    out[0] = lsum / (2.0f * Bf);                 // loss
    out[1] = psum / Bf;                          // mean_pos
    out[2] = nsum / (Bf * (Bf - 1.0f));          // mean_neg
    out[3] = (accA + accB) / (2.0f * Bf);        // accuracy
  }
}

// ---------------------------------------------------------------------------
extern "C" void kernel_launch(void* const* d_in, const int* in_sizes, int n_in,
                              void* d_out, int out_size, void* d_ws, size_t ws_size,
                              hipStream_t stream) {
  const float* a = (const float*)d_in[0];
  const float* b = (const float*)d_in[1];
  float* out = (float*)d_out;

  // Workspace layout: 6 f32[16384] arrays, then 2 u64[16384] (offset is 8B-aligned).
  float* inva  = (float*)d_ws;
  float* invb  = inva + B_ROWS;
  float* diagv = invb + B_ROWS;
  float* rowExp = diagv + B_ROWS;
  float* rowSum = rowExp + B_ROWS;
  float* colExp = rowSum + B_ROWS;
  unsigned long long* rowMax = (unsigned long long*)(colExp + B_ROWS);
  unsigned long long* colMax = rowMax + B_ROWS;

  init_acc_kernel<<<B_ROWS / 256, 256, 0, stream>>>(rowExp, rowSum, colExp, rowMax, colMax);
  inv_norm_kernel<<<dim3(B_ROWS, 2), 256, 0, stream>>>(a, b, inva, invb);
  sim_gemm_reduce_kernel<<<dim3(B_ROWS / TN, B_ROWS / TM), 256, 0, stream>>>(
      a, b, inva, invb, rowExp, rowSum, colExp, rowMax, colMax, diagv);
  finalize_kernel<<<1, 256, 0, stream>>>(rowExp, rowSum, colExp, rowMax, colMax, diagv, out);
}